// DiceCoeffMetric_3255585211067
// MI455X (gfx1250) — compile-verified
//
#include <hip/hip_runtime.h>

#define EPS 1e-6f

typedef __attribute__((ext_vector_type(2))) float v2f;
typedef __attribute__((ext_vector_type(4))) float v4f;
typedef __attribute__((ext_vector_type(8))) float v8f;

constexpr int kB            = 16;
constexpr int kC            = 4;
constexpr int kPlane        = 1024 * 1024;          // H*W elements per (b,c) plane
constexpr int kBlockThreads = 256;                  // 8 waves (wave32)
constexpr int kChunk        = 65536;                // elements per block (contiguous, within one plane)
constexpr int kBlocksPerPlane = kPlane / kChunk;    // 16
constexpr int kNumPlanes    = kB * kC;              // 64
constexpr int kGrid         = kNumPlanes * kBlocksPerPlane;   // 1024 blocks
constexpr int kIters        = kChunk / (kBlockThreads * 4);   // 64 b128 loads per thread

// Stage 1: streaming reduction. Each wave accumulates sum(prd*tgt) into a
// 16x16 f32 WMMA accumulator by multiplying against an all-ones B matrix:
//   D[m][n] += sum_k A[m][k] * 1
// Every column of D holds the same row-sums, so the wave-total is
// (sum over lanes of sum over the 8 C VGPRs) / 16.  Sum(prd) and sum(tgt)
// are plain VALU adds. All loads are non-temporal b128 (512 MiB streamed once,
// no reuse -> don't pollute L2).
__global__ __launch_bounds__(kBlockThreads)
void dice_partial_kernel(const float* __restrict__ prd,
                         const float* __restrict__ tgt,
                         float* __restrict__ partials)
{
    const int tid = threadIdx.x;
    const int bid = blockIdx.x;

    const v4f* __restrict__ p4 = reinterpret_cast<const v4f*>(prd) + (size_t)bid * (kChunk / 4);
    const v4f* __restrict__ t4 = reinterpret_cast<const v4f*>(tgt) + (size_t)bid * (kChunk / 4);

    v8f acc = {};                      // wave-cooperative 16x16 f32 C/D matrix
    const v2f ones = {1.0f, 1.0f};     // all-ones B operand (layout-independent)
    float sp = 0.0f, st = 0.0f;

#pragma unroll 4
    for (int j = 0; j < kIters; ++j) {
        v4f a = __builtin_nontemporal_load(&p4[j * kBlockThreads + tid]);
        v4f b = __builtin_nontemporal_load(&t4[j * kBlockThreads + tid]);

        v2f q0 = { a.x * b.x, a.y * b.y };
        v2f q1 = { a.z * b.z, a.w * b.w };

        // 8-arg form: (neg_a, A, neg_b, B, c_mod, C, reuse_a, reuse_b)
        acc = __builtin_amdgcn_wmma_f32_16x16x4_f32(
                  false, q0, false, ones, (short)0, acc, false, false);
        acc = __builtin_amdgcn_wmma_f32_16x16x4_f32(
                  false, q1, false, ones, (short)0, acc, false, false);

        sp += (a.x + a.y) + (a.z + a.w);
        st += (b.x + b.y) + (b.z + b.w);
    }

    // Per-lane partial of the replicated dot-product accumulator.
    float di = ((acc[0] + acc[1]) + (acc[2] + acc[3])) +
               ((acc[4] + acc[5]) + (acc[6] + acc[7]));

    // wave32 tree reductions (fixed order -> deterministic)
    for (int off = 16; off > 0; off >>= 1) {
        di += __shfl_down(di, off, 32);
        sp += __shfl_down(sp, off, 32);
        st += __shfl_down(st, off, 32);
    }
    di *= 0.0625f;   // each row-sum appears in 16 columns; undo exact x16

    __shared__ float s_red[3][kBlockThreads / 32];
    const int wave = tid >> 5;
    if ((tid & 31) == 0) {
        s_red[0][wave] = di;
        s_red[1][wave] = sp;
        s_red[2][wave] = st;
    }
    __syncthreads();

    if (tid == 0) {
        float I = 0.0f, P = 0.0f, T = 0.0f;
        for (int w = 0; w < kBlockThreads / 32; ++w) {
            I += s_red[0][w];
            P += s_red[1][w];
            T += s_red[2][w];
        }
        partials[bid * 3 + 0] = I;
        partials[bid * 3 + 1] = P;
        partials[bid * 3 + 2] = T;
    }
}

// Stage 2: one block, one thread per (b,c) plane. Deterministic serial
// combine of the 16 per-block partials, dice, then per-channel batch mean.
__global__ __launch_bounds__(kNumPlanes)
void dice_finalize_kernel(const float* __restrict__ partials,
                          float* __restrict__ out)
{
    __shared__ float sd[kNumPlanes];
    const int p = threadIdx.x;           // plane = b*C + c

    float I = 0.0f, P = 0.0f, T = 0.0f;
    for (int s = 0; s < kBlocksPerPlane; ++s) {
        const int base = (p * kBlocksPerPlane + s) * 3;
        I += partials[base + 0];
        P += partials[base + 1];
        T += partials[base + 2];
    }
    sd[p] = (2.0f * I + EPS) / (P + T + EPS);
    __syncthreads();

    if (p < kC) {
        float m = 0.0f;
        for (int b = 0; b < kB; ++b) m += sd[b * kC + p];
        out[p] = m * (1.0f / (float)kB);
    }
}

extern "C" void kernel_launch(void* const* d_in, const int* in_sizes, int n_in,
                              void* d_out, int out_size, void* d_ws, size_t ws_size,
                              hipStream_t stream)
{
    const float* prd = (const float*)d_in[0];
    const float* tgt = (const float*)d_in[1];
    float* ws  = (float*)d_ws;   // 1024 blocks * 3 floats = 12 KiB
    float* out = (float*)d_out;  // (C,) = 4 floats

    dice_partial_kernel<<<kGrid, kBlockThreads, 0, stream>>>(prd, tgt, ws);
    dice_finalize_kernel<<<1, kNumPlanes, 0, stream>>>(ws, out);
}